// SolvePointQP_48180943126988
// MI455X (gfx1250) — compile-verified
//
#include <hip/hip_runtime.h>
#include <stdint.h>

// ---------------------------------------------------------------------------
// Batched QP interior-point solver, restructured for MI455X (gfx1250).
//
// Per batch element: min 0.5 z'Qz + p'z  s.t. Gz + s = h, s,lam >= 0
//   z = [x(24), u(24), v(24)], Q = 1e-3 I, G has <=2 nonzeros/row.
// KKT = Q + G' diag(w) G has structure:
//   [ T(tridiag)  diag(a)  -diag(b) ]
//   [ diag(a)     diag(du)    0     ]
//   [ -diag(b)      0      diag(dv) ]
// Eliminate u,v -> strictly diagonally dominant SPD 24x24 tridiagonal system
// solved by the Thomas algorithm. One wave32 per batch element.
// ---------------------------------------------------------------------------

#define N_    24
#define D_    72
#define M_    166          // n + n + 2(n-1) + 3n rows (matches G/h as built)
#define NIT   25
#define SIGMA 0.1f
#define FRAC  0.99f
#define CRAMP 0.4f
#define GU    50.0f
#define GO    0.5f
#define BIG   1e12f

__device__ __forceinline__ float rcpf(float x) { return __builtin_amdgcn_rcpf(x); }

__device__ __forceinline__ float wave_sum(float v) {
#pragma unroll
  for (int off = 16; off > 0; off >>= 1) v += __shfl_xor(v, off, 32);
  return v;
}
__device__ __forceinline__ float wave_min(float v) {
#pragma unroll
  for (int off = 16; off > 0; off >>= 1) v = fminf(v, __shfl_xor(v, off, 32));
  return v;
}

#if defined(__gfx1250__) && __has_builtin(__builtin_amdgcn_global_load_async_to_lds_b32) && __has_builtin(__builtin_amdgcn_s_wait_asynccnt)
#define USE_ASYNC_LDS 1
#else
#define USE_ASYNC_LDS 0
#endif

__global__ __launch_bounds__(32)
void qp_ipm_tridiag_kernel(const float* __restrict__ pred,
                           float* __restrict__ out) {
  __shared__ float predL[N_];
  __shared__ float zL[D_];
  __shared__ float ybuf[M_];   // holds lam, then rv (reused)
  __shared__ float wbuf[M_];   // constraint weights w = lam/s
  __shared__ float triD[N_], triR[N_ - 1], triB[N_];
  __shared__ float dzL[D_];

  const int lane = threadIdx.x;
  const int b = blockIdx.x;
  const float* predB = pred + b * N_;

  // ---- stage pred[b][0..23] into LDS (async DMA path on gfx1250) ----
#if USE_ASYNC_LDS
  if (lane < N_) {
    __builtin_amdgcn_global_load_async_to_lds_b32(
        (__attribute__((address_space(1))) int*)(predB + lane),
        (__attribute__((address_space(3))) int*)(&predL[lane]),
        /*imm_offset=*/0, /*cpol=*/0);
  }
#else
  if (lane < N_) predL[lane] = predB[lane];
#endif

  // init z = 0 (lanes 0..23 own columns lane, 24+lane, 48+lane)
  if (lane < N_) {
    zL[lane] = 0.f; zL[24 + lane] = 0.f; zL[48 + lane] = 0.f;
  }

#if USE_ASYNC_LDS
  __builtin_amdgcn_s_wait_asynccnt(0);
#endif
  __syncthreads();

  // per-lane constraint slots: row r = lane + 32k, k = 0..5 (valid if r < 166)
  float s[6], lm[6], hh[6];
#pragma unroll
  for (int k = 0; k < 6; ++k) { s[k] = 1.f; lm[k] = 1.f; }
#pragma unroll
  for (int k = 0; k < 6; ++k) {
    int r = lane + 32 * k;
    float hv = 0.f;
    if (r < 24)       hv = -predL[r];
    else if (r < 48)  hv = predL[r - 24];
    else if (r < 94)  hv = CRAMP;
    hh[k] = hv;
  }

  for (int it = 0; it < NIT; ++it) {
    // ---- 1: publish lam for G'lam gathers ----
#pragma unroll
    for (int k = 0; k < 6; ++k) { int r = lane + 32 * k; if (r < M_) ybuf[r] = lm[k]; }
    __syncthreads();

    // ---- 2: dual residual r_d (lanes<24), primal residual r_p, mu, w, rv ----
    float rdx = 0.f, rdu = 0.f, rdv = 0.f;
    if (lane < N_) {
      int j = lane;
      float gx = -ybuf[j] + ybuf[24 + j] - ybuf[94 + j];
      if (j <= 22) gx += ybuf[48 + 2 * j] - ybuf[49 + 2 * j];
      if (j >= 1)  gx += -ybuf[48 + 2 * j - 2] + ybuf[49 + 2 * j - 2];
      float gu = -ybuf[j] - ybuf[118 + j];
      float gv = -ybuf[24 + j] - ybuf[142 + j];
      rdx = 1e-3f * zL[j]       + gx;
      rdu = 1e-3f * zL[24 + j]  + GU + gu;
      rdv = 1e-3f * zL[48 + j]  + GO + gv;
    }

    float rp[6], w[6], rv[6];
    float part = 0.f;
#pragma unroll
    for (int k = 0; k < 6; ++k) {
      int r = lane + 32 * k;
      if (r < M_) {
        float gz;
        if (r < 24)       gz = -zL[r] - zL[24 + r];
        else if (r < 48)  gz = zL[r - 24] - zL[r + 24];
        else if (r < 94)  { int t = r - 48, i = t >> 1; float v = zL[i] - zL[i + 1]; gz = (t & 1) ? -v : v; }
        else              gz = -zL[r - 94];
        rp[k] = gz + s[k] - hh[k];
        part += s[k] * lm[k];
      } else { rp[k] = 0.f; }
    }
    float mu = wave_sum(part) * (1.0f / (float)M_);
#pragma unroll
    for (int k = 0; k < 6; ++k) {
      int r = lane + 32 * k;
      if (r < M_) {
        float is = rcpf(s[k]);
        w[k]  = lm[k] * is;
        rv[k] = (SIGMA * mu - s[k] * lm[k]) * is + w[k] * rp[k];
      } else { w[k] = 0.f; rv[k] = 0.f; }
    }
    __syncthreads();   // all lam reads done before ybuf is reused

    // ---- 3: publish w and rv ----
#pragma unroll
    for (int k = 0; k < 6; ++k) {
      int r = lane + 32 * k;
      if (r < M_) { wbuf[r] = w[k]; ybuf[r] = rv[k]; }
    }
    __syncthreads();

    // ---- 4: rhs = -(r_d + G'rv); Schur-reduced tridiagonal coefficients ----
    float a = 0.f, bb = 0.f, idu = 0.f, idv = 0.f, rhu = 0.f, rhv = 0.f;
    if (lane < N_) {
      int j = lane;
      float gx = -ybuf[j] + ybuf[24 + j] - ybuf[94 + j];
      if (j <= 22) gx += ybuf[48 + 2 * j] - ybuf[49 + 2 * j];
      if (j >= 1)  gx += -ybuf[48 + 2 * j - 2] + ybuf[49 + 2 * j - 2];
      float gu = -ybuf[j] - ybuf[118 + j];
      float gv = -ybuf[24 + j] - ybuf[142 + j];
      float rhx = -(rdx + gx);
      rhu = -(rdu + gu);
      rhv = -(rdv + gv);

      a  = wbuf[j];
      bb = wbuf[24 + j];
      float fx = wbuf[94 + j], fu = wbuf[118 + j], fv = wbuf[142 + j];
      float rprev = (j >= 1)  ? (wbuf[48 + 2 * j - 2] + wbuf[49 + 2 * j - 2]) : 0.f;
      float rnext = (j <= 22) ? (wbuf[48 + 2 * j]     + wbuf[49 + 2 * j])     : 0.f;
      idu = rcpf(1e-3f + a  + fu);
      idv = rcpf(1e-3f + bb + fv);
      float dgx = 1e-3f + a + bb + fx + rprev + rnext;
      triD[j] = dgx - a * a * idu - bb * bb * idv;   // Schur diagonal
      if (j <= 22) triR[j] = rnext;                  // off-diag magnitude (entry = -rnext)
      triB[j] = rhx - a * idu * rhu + bb * idv * rhv;
    }
    __syncthreads();

    // ---- 5: Thomas solve of 24x24 SPD tridiagonal (redundant on all lanes,
    //         fully unrolled -> register arrays; lane 0 publishes) ----
    float Dv[N_], Bv[N_], Rv[N_ - 1], iD[N_];
#pragma unroll
    for (int i = 0; i < N_; ++i) { Dv[i] = triD[i]; Bv[i] = triB[i]; }
#pragma unroll
    for (int i = 0; i < N_ - 1; ++i) Rv[i] = triR[i];
    iD[0] = rcpf(Dv[0]);
#pragma unroll
    for (int i = 1; i < N_; ++i) {
      float fac = -Rv[i - 1] * iD[i - 1];   // c/D', c = -Rv
      Dv[i] += fac * Rv[i - 1];             // D - fac*c
      Bv[i] -= fac * Bv[i - 1];
      iD[i] = rcpf(Dv[i]);
    }
    float X[N_];
    X[N_ - 1] = Bv[N_ - 1] * iD[N_ - 1];
#pragma unroll
    for (int i = N_ - 2; i >= 0; --i) X[i] = (Bv[i] + Rv[i] * X[i + 1]) * iD[i];
    if (lane == 0) {
#pragma unroll
      for (int i = 0; i < N_; ++i) dzL[i] = X[i];
    }
    __syncthreads();

    // ---- 6: back out dz_u, dz_v ----
    if (lane < N_) {
      float dzx = dzL[lane];
      dzL[24 + lane] = (rhu - a  * dzx) * idu;
      dzL[48 + lane] = (rhv + bb * dzx) * idv;
    }
    __syncthreads();

    // ---- 7: ds, dlam, fraction-to-boundary alpha, update ----
    float amin = BIG;
    float dsv[6], dlv[6];
#pragma unroll
    for (int k = 0; k < 6; ++k) {
      int r = lane + 32 * k;
      if (r < M_) {
        float gdz;
        if (r < 24)       gdz = -dzL[r] - dzL[24 + r];
        else if (r < 48)  gdz = dzL[r - 24] - dzL[r + 24];
        else if (r < 94)  { int t = r - 48, i = t >> 1; float v = dzL[i] - dzL[i + 1]; gdz = (t & 1) ? -v : v; }
        else              gdz = -dzL[r - 94];
        float ds = -rp[k] - gdz;
        float is = rcpf(s[k]);
        float dl = (SIGMA * mu - s[k] * lm[k] - lm[k] * ds) * is;
        dsv[k] = ds; dlv[k] = dl;
        float c1 = (ds < 0.f) ? (-s[k]  * rcpf(ds)) : BIG;
        float c2 = (dl < 0.f) ? (-lm[k] * rcpf(dl)) : BIG;
        amin = fminf(amin, fminf(c1, c2));
      } else { dsv[k] = 0.f; dlv[k] = 0.f; }
    }
    amin = wave_min(amin);
    float alpha = fminf(1.0f, FRAC * amin);
#pragma unroll
    for (int k = 0; k < 6; ++k) {
      int r = lane + 32 * k;
      if (r < M_) { s[k] += alpha * dsv[k]; lm[k] += alpha * dlv[k]; }
    }
    if (lane < N_) {
      zL[lane]      += alpha * dzL[lane];
      zL[24 + lane] += alpha * dzL[24 + lane];
      zL[48 + lane] += alpha * dzL[48 + lane];
    }
    __syncthreads();
  }

  // ---- write z (B, 72) ----
  if (lane < N_) {
    float* ob = out + b * D_;
    ob[lane]      = zL[lane];
    ob[24 + lane] = zL[24 + lane];
    ob[48 + lane] = zL[48 + lane];
  }
}

extern "C" void kernel_launch(void* const* d_in, const int* in_sizes, int n_in,
                              void* d_out, int out_size, void* d_ws, size_t ws_size,
                              hipStream_t stream) {
  (void)n_in; (void)out_size; (void)d_ws; (void)ws_size;
  const float* pred = (const float*)d_in[0];
  float* out = (float*)d_out;
  const int B = in_sizes[0] / N_;   // 2048 for the reference shapes
  qp_ipm_tridiag_kernel<<<dim3(B), dim3(32), 0, stream>>>(pred, out);
}